// EctLayer_56203942035574
// MI455X (gfx1250) — compile-verified
//
#include <hip/hip_runtime.h>

typedef __attribute__((ext_vector_type(2))) float v2f;
typedef __attribute__((ext_vector_type(8))) float v8f;

#define TDIR 32            // T: directions (columns of v)
#define BTH  32            // B: thresholds
#define CELLS (TDIR * BTH) // 1024 cells per graph
#define CHUNK 128          // nodes/edges per accumulation block

// 200 * log2(e): sigmoid(200*(lin-h)) = 1/(1+exp2(K*h - K*lin))
#define KSIG 288.53900817779268f

// ---------------------------------------------------------------- zero fill
__global__ void k_zero(float* __restrict__ p, int n) {
  int i = blockIdx.x * blockDim.x + threadIdx.x;
  if (i < n) p[i] = 0.0f;
}

// ------------------------------------------------- nh = x(N,3) @ v(3,32), WMMA
// V_WMMA_F32_16X16X4_F32, K padded 3 -> 4 with zeros.
// A 16x4 f32 layout (ISA 7.12.2): lanes 0-15 hold M=lane, VGPR0=K0, VGPR1=K1;
//                                 lanes 16-31 hold M=lane-16, VGPR0=K2, VGPR1=K3.
// B 4x16 f32: mirrored (N across lanes, K across VGPRs / lane halves).
// D 16x16 f32: VGPR r: lanes 0-15 -> M=r, N=lane; lanes 16-31 -> M=r+8, N=lane-16.
// nh is padded to a multiple of 16 rows in workspace -> unguarded stores,
// x-row loads clamped to N-1 (pad rows computed but never read back).
__global__ void k_nh_wmma(const float* __restrict__ x, const float* __restrict__ v,
                          float* __restrict__ nh, int N) {
  int wave = blockIdx.x * (blockDim.x >> 5) + (threadIdx.x >> 5);
  int lane = threadIdx.x & 31;
  int m0 = wave << 4;               // 16 nodes per wave
  if (m0 >= N) return;              // wave-uniform exit keeps EXEC all-ones

  bool hi = lane >= 16;
  int mn = lane & 15;

  // A tile (16 nodes x K4), branchless: clamp row for load, select pad zeros
  int node = m0 + mn;
  int nc = node < N ? node : N - 1;
  float a0 = x[nc * 3 + (hi ? 2 : 0)];       // lanes<16: K0, lanes>=16: K2
  float a1 = hi ? 0.0f : x[nc * 3 + 1];      // lanes<16: K1, lanes>=16: K3=pad
  v2f A = {a0, a1};

  // B tiles (K4 x 16 dirs), dirs 0-15 and 16-31
  int krow = hi ? 2 : 0;
  float b0a = v[krow * TDIR + mn];
  float b0b = hi ? 0.0f : v[1 * TDIR + mn];
  float b1a = v[krow * TDIR + 16 + mn];
  float b1b = hi ? 0.0f : v[1 * TDIR + 16 + mn];
  v2f B0 = {b0a, b0b};
  v2f B1 = {b1a, b1b};

  v8f c0 = {};
  v8f c1 = {};
  // (neg_a, A, neg_b, B, c_mod, C, reuse_a, reuse_b)
  c0 = __builtin_amdgcn_wmma_f32_16x16x4_f32(false, A, false, B0, (short)0, c0, false, false);
  c1 = __builtin_amdgcn_wmma_f32_16x16x4_f32(false, A, false, B1, (short)0, c1, false, false);

  int mbase = m0 + (hi ? 8 : 0);
#pragma unroll
  for (int r = 0; r < 8; ++r) {
    int nd = mbase + r;                     // unguarded: nh padded to 16 rows
    nh[nd * TDIR + mn]      = c0[r];
    nh[nd * TDIR + 16 + mn] = c1[r];
  }
}

// ----------------------------------------------- edge binning by graph id
__global__ void k_count(const int* __restrict__ ei, const int* __restrict__ batch,
                        int* __restrict__ counts, int E) {
  int e = blockIdx.x * blockDim.x + threadIdx.x;
  if (e < E) atomicAdd(&counts[batch[ei[e]]], 1);
}

__global__ void k_scan(const int* __restrict__ counts, int* __restrict__ cursor, int G) {
  if (blockIdx.x == 0 && threadIdx.x == 0) {
    int s = 0;
    for (int g = 0; g < G; ++g) { cursor[g] = s; s += counts[g]; }
  }
}

__global__ void k_scatter(const int* __restrict__ ei, const int* __restrict__ batch,
                          int* __restrict__ cursor, int* __restrict__ binned, int E) {
  int e = blockIdx.x * blockDim.x + threadIdx.x;
  if (e < E) {
    int g = batch[ei[e]];
    int pos = atomicAdd(&cursor[g], 1);
    binned[pos] = e;
  }
}

// sigmoid(200*(lin-h)) with c = -K*lin precomputed:
// fma + v_exp_f32 + add + v_rcp_f32  (saturates correctly at 0/1)
__device__ __forceinline__ float sig_fast(float h, float c) {
  return __builtin_amdgcn_rcpf(1.0f + __builtin_amdgcn_exp2f(__builtin_fmaf(h, KSIG, c)));
}

// ------------------------------------------ node accumulation (sorted batch)
// 1024 threads/block: thread = (b,t) cell. Register accumulate across a
// 128-node chunk; flush with one float atomic per cell per graph segment.
__global__ void k_nodes(const float* __restrict__ nh, const int* __restrict__ batch,
                        const float* __restrict__ lin, float* __restrict__ out, int N) {
  __shared__ int s_g[CHUNK];
  int tid = threadIdx.x;
  int b = tid >> 5, t = tid & 31;
  int base = blockIdx.x * CHUNK;
  int cnt = N - base; if (cnt > CHUNK) cnt = CHUNK;
  for (int i = tid; i < cnt; i += blockDim.x) s_g[i] = batch[base + i];
  __syncthreads();
  if (cnt <= 0) return;

  float c = -KSIG * lin[b];
  float acc = 0.f;
  int cur = s_g[0];
#pragma unroll 4
  for (int i = 0; i < cnt; ++i) {
    int g = s_g[i];                       // block-uniform
    if (g != cur) {                       // uniform branch, rare
      atomicAdd(&out[cur * CELLS + tid], acc);
      acc = 0.f; cur = g;
    }
    float h = nh[(size_t)(base + i) * TDIR + t];   // coalesced 128B row / wave
    acc += sig_fast(h, c);
  }
  atomicAdd(&out[cur * CELLS + tid], acc);
}

// ---------------------------------- edge accumulation (graph-binned edges)
// Accumulates with negative sign: out = ect_nodes - ect_edges fused.
__global__ void k_edges(const float* __restrict__ nh, const int* __restrict__ batch,
                        const int* __restrict__ ei, const int* __restrict__ binned,
                        const float* __restrict__ lin, float* __restrict__ out, int E) {
  __shared__ int s_u[CHUNK], s_w[CHUNK], s_g[CHUNK];
  int tid = threadIdx.x;
  int b = tid >> 5, t = tid & 31;
  int base = blockIdx.x * CHUNK;
  int cnt = E - base; if (cnt > CHUNK) cnt = CHUNK;
  for (int i = tid; i < cnt; i += blockDim.x) {
    int e = binned[base + i];
    int u = ei[e];
    int w = ei[E + e];
    s_u[i] = u; s_w[i] = w; s_g[i] = batch[u];
  }
  __syncthreads();
  if (cnt <= 0) return;

  float c = -KSIG * lin[b];
  float acc = 0.f;
  int cur = s_g[0];
#pragma unroll 4
  for (int i = 0; i < cnt; ++i) {
    int g = s_g[i];                       // sorted via binning -> few flushes
    if (g != cur) {
      atomicAdd(&out[cur * CELLS + tid], -acc);
      acc = 0.f; cur = g;
    }
    float hu = nh[(size_t)s_u[i] * TDIR + t];
    float hw = nh[(size_t)s_w[i] * TDIR + t];
    float h = fmaxf(hu, hw);              // eh = max over the two endpoints
    acc += sig_fast(h, c);
  }
  atomicAdd(&out[cur * CELLS + tid], -acc);
}

// ---------------------------------------------------------------- launcher
extern "C" void kernel_launch(void* const* d_in, const int* in_sizes, int n_in,
                              void* d_out, int out_size, void* d_ws, size_t ws_size,
                              hipStream_t stream) {
  const float* x     = (const float*)d_in[0];   // (N,3)
  const float* v     = (const float*)d_in[1];   // (3,32)
  const float* lin   = (const float*)d_in[2];   // (32,)
  const int*   ei    = (const int*)d_in[3];     // (2,E) flat
  const int*   batch = (const int*)d_in[4];     // (N,) sorted
  int N = in_sizes[0] / 3;
  int E = in_sizes[3] / 2;
  int G = out_size / CELLS;
  float* out = (float*)d_out;

  int tiles = (N + 15) / 16;
  int Npad = tiles * 16;                         // padded rows for unguarded WMMA stores

  char* ws = (char*)d_ws;
  float* nh = (float*)ws;                        // Npad*32 f32
  size_t off = (size_t)Npad * TDIR * sizeof(float);
  int* counts = (int*)(ws + off); off += (size_t)G * sizeof(int);
  int* cursor = (int*)(ws + off); off += (size_t)G * sizeof(int);
  int* binned = (int*)(ws + off);                // E ints

  // zero output and counts+cursor (contiguous)
  k_zero<<<(out_size + 255) / 256, 256, 0, stream>>>(out, out_size);
  k_zero<<<(2 * G + 255) / 256, 256, 0, stream>>>((float*)counts, 2 * G);

  // WMMA GEMM: nh = x @ v
  k_nh_wmma<<<(tiles + 7) / 8, 256, 0, stream>>>(x, v, nh, N);

  // bin edges by graph
  k_count  <<<(E + 255) / 256, 256, 0, stream>>>(ei, batch, counts, E);
  k_scan   <<<1, 32, 0, stream>>>(counts, cursor, G);
  k_scatter<<<(E + 255) / 256, 256, 0, stream>>>(ei, batch, cursor, binned, E);

  // sigmoid histogram accumulation (nodes +, edges -)
  k_nodes<<<(N + CHUNK - 1) / CHUNK, CELLS, 0, stream>>>(nh, batch, lin, out, N);
  k_edges<<<(E + CHUNK - 1) / CHUNK, CELLS, 0, stream>>>(nh, batch, ei, binned, lin, out, E);
}